// SpanRepLayer_51041391345719
// MI455X (gfx1250) — compile-verified
//
#include <hip/hip_runtime.h>
#include <stdint.h>

// Problem constants (match reference: B,K,S,H = 8,2048,1024,256; W=24; WIN=3)
#define BB   8
#define KK   2048
#define SS   1024
#define HH   256
#define WW   24
#define WIN  3

#define THREADS 256
#define NSPAN   8                                   // spans per block (pipelined)
#define WFLOATS (WW * HH)                           // 6144 floats = 24 KB window
#define CHUNKS  ((WFLOATS * 4) / (THREADS * 16))    // 6 x b128 async issues per thread

// Async DMA (global -> LDS) of the first `jmax` 4-row chunks of a span window.
// Each chunk = 256 threads x 16B = 4 KB = 4 rows of H=256 f32.
__device__ __forceinline__ void issue_window_async(const float* tok, int b, int st,
                                                   int jmax, float* lds_buf, int tid)
{
    const float* g = tok + ((size_t)b * SS + (size_t)st) * HH;
    uint64_t ga = (uint64_t)(uintptr_t)g + (uint64_t)(tid * 16);
    // Low 32 bits of a generic pointer into the LDS aperture are the LDS byte
    // address (ISA 10.2).
    uint32_t la = (uint32_t)(uintptr_t)lds_buf + (uint32_t)(tid * 16);
#pragma unroll
    for (int j = 0; j < CHUNKS; ++j) {
        if (j < jmax) {                         // uniform guard: skip rows >= width
            uint32_t l  = la + (uint32_t)(j * THREADS * 16);
            uint64_t gj = ga + (uint64_t)(j * THREADS * 16);
            asm volatile("global_load_async_to_lds_b128 %0, %1, off"
                         :: "v"(l), "v"(gj)
                         : "memory");
        }
    }
}

__global__ __launch_bounds__(THREADS)
void span_rep_kernel(const float*   __restrict__ tok,    // (B,S,H) f32
                     const int*     __restrict__ ids,    // (B,K,2) i32
                     const uint8_t* __restrict__ masks,  // (B,K)  bool (1 byte)
                     float*         __restrict__ out)    // (B,K,3H) f32
{
    __shared__ float buf[2][WFLOATS];   // 48 KB double buffer
    const int tid   = threadIdx.x;      // channel h = tid
    const int span0 = blockIdx.x * NSPAN;
    const int b     = span0 >> 11;      // span0 / KK; all NSPAN spans share batch

    // Prologue: hoist all block-uniform span metadata into SGPR-resident values.
    int st[NSPAN], wd[NSPAN];
    uint32_t mk[NSPAN];
#pragma unroll
    for (int i = 0; i < NSPAN; ++i) {
        const int s = ids[2 * (span0 + i)];
        const int e = ids[2 * (span0 + i) + 1];
        st[i] = s;
        wd[i] = e - s;                 // 1..24
        mk[i] = masks[span0 + i];
    }

    // Start DMA of the first span's (truncated) window.
    issue_window_async(tok, b, st[0], (wd[0] + 3) >> 2, buf[0], tid);

#pragma unroll
    for (int i = 0; i < NSPAN; ++i) {
        // Wait for this wave's async DMA, then make all waves' portions visible.
        asm volatile("s_wait_asynccnt 0" ::: "memory");
        __syncthreads();

        // Kick off next span's DMA into the other buffer; streams from L2 while
        // we reduce the current buffer.
        if (i + 1 < NSPAN)
            issue_window_async(tok, b, st[i + 1], (wd[i + 1] + 3) >> 2,
                               buf[(i + 1) & 1], tid);

        const int w = wd[i];
        const float* __restrict__ L = buf[i & 1] + tid;
        const float NEG = -__builtin_huge_valf();

        // Three pools over exactly the live ranges (all bounds block-uniform,
        // so loop control is pure scalar work; ds reads are conflict-free b32).
        float smax = NEG, emax = NEG;
        const int s_hi = (w < WIN) ? w : WIN;           // [0, min(3,w))
        for (int t = 0; t < s_hi; ++t)
            smax = fmaxf(smax, L[t * HH]);

        const int e_lo = (w > WIN) ? (w - WIN) : 0;     // [max(w-3,0), w)
        for (int t = e_lo; t < w; ++t)
            emax = fmaxf(emax, L[t * HH]);

        float inner = smax;                             // fallback when w <= 6
        if (w > 2 * WIN) {                              // [3, w-3)
            float imax = NEG;
            for (int t = WIN; t < w - WIN; ++t)
                imax = fmaxf(imax, L[t * HH]);
            inner = imax;
        }

        const bool valid = mk[i] != 0;
        const size_t o = (size_t)(span0 + i) * (3 * HH) + (size_t)tid;
        out[o]          = valid ? smax  : 0.0f;
        out[o + HH]     = valid ? inner : 0.0f;
        out[o + 2 * HH] = valid ? emax  : 0.0f;
        // Buffer-reuse ordering for iteration i+2 is enforced by the top-of-loop
        // wait + barrier of iteration i+1.
    }
}

extern "C" void kernel_launch(void* const* d_in, const int* in_sizes, int n_in,
                              void* d_out, int out_size, void* d_ws, size_t ws_size,
                              hipStream_t stream)
{
    (void)in_sizes; (void)n_in; (void)out_size; (void)d_ws; (void)ws_size;
    const float*   tok   = (const float*)d_in[0];
    const int*     ids   = (const int*)d_in[1];
    const uint8_t* masks = (const uint8_t*)d_in[2];
    float*         out   = (float*)d_out;

    const int nblocks = (BB * KK) / NSPAN;   // 2048 blocks
    span_rep_kernel<<<nblocks, THREADS, 0, stream>>>(tok, ids, masks, out);
}